// Net_37108517437448
// MI455X (gfx1250) — compile-verified
//
#include <hip/hip_runtime.h>
#include <math.h>

typedef __attribute__((ext_vector_type(2))) float v2f;
typedef __attribute__((ext_vector_type(8))) float v8f;

#define D_IN  128
#define D_HID 16
#define D_OUT 2

// ---------------------------------------------------------------------------
// Kernel 1: deg[i] = 1.0 (self loop)
__global__ void k_deg_init(float* __restrict__ deg, int n) {
    int i = blockIdx.x * blockDim.x + threadIdx.x;
    if (i < n) deg[i] = 1.0f;
}

// Kernel 2: deg[dst[e]] += 1
__global__ void k_deg_accum(const int* __restrict__ dst, float* __restrict__ deg, int e) {
    int i = blockIdx.x * blockDim.x + threadIdx.x;
    if (i < e) atomicAdd(&deg[dst[i]], 1.0f);
}

// Kernel 3: dinv[i] = rsqrt(deg[i])   (in place; deg>=1 always)
__global__ void k_deg_rsqrt(float* __restrict__ deg, int n) {
    int i = blockIdx.x * blockDim.x + threadIdx.x;
    if (i < n) deg[i] = rsqrtf(deg[i]);
}

// ---------------------------------------------------------------------------
// Kernel 4: h = x @ W1 via V_WMMA_F32_16X16X4_F32.  One wave per 16-row tile.
//   hs[row]   = h[row] * dinv[row]                (pre-scaled messages)
//   agg1[row] = hs[row] * dinv[row]               (self-loop contribution seed)
__global__ void k_gemm1_wmma(const float* __restrict__ x,
                             const float* __restrict__ W1,
                             const float* __restrict__ dinv,
                             float* __restrict__ hs,
                             float* __restrict__ agg1,
                             int ntiles) {
    int wave = blockIdx.x * (blockDim.x >> 5) + (threadIdx.x >> 5);
    if (wave >= ntiles) return;              // wave-uniform -> EXEC all-1s inside
    int lane  = threadIdx.x & 31;
    int col   = lane & 15;                   // N index (and M index for A loads)
    int khalf = (lane >> 4) << 1;            // 0 for lanes 0-15, 2 for lanes 16-31
    int row0  = wave << 4;

    const float* xrow = x + (size_t)(row0 + col) * D_IN;  // A: M = lane&15

    v8f c = {};
    #pragma unroll
    for (int k0 = 0; k0 < D_IN; k0 += 4) {
        v2f a, b;
        // A fragment (16x4 f32): VGPR0=K{k0+khalf}, VGPR1=K{k0+khalf+1}
        a.x = xrow[k0 + khalf];
        a.y = xrow[k0 + khalf + 1];
        // B fragment (4x16 f32): lanes hold column N=col, VGPRs hold K
        b.x = W1[(k0 + khalf) * D_HID + col];
        b.y = W1[(k0 + khalf + 1) * D_HID + col];
        c = __builtin_amdgcn_wmma_f32_16x16x4_f32(
                /*neg_a=*/false, a, /*neg_b=*/false, b,
                /*c_mod=*/(short)0, c, /*reuse_a=*/false, /*reuse_b=*/false);
    }

    // C/D layout: VGPR r, lanes 0-15 -> M=r, lanes 16-31 -> M=r+8; N = lane&15
    int rbase = row0 + ((lane >> 4) << 3);
    #pragma unroll
    for (int r = 0; r < 8; ++r) {
        int row  = rbase + r;
        float di = dinv[row];
        float hv = c[r] * di;                 // hs = h * dinv[row]
        hs[(size_t)row * D_HID + col]   = hv;
        agg1[(size_t)row * D_HID + col] = hv * di;   // self-loop seed
    }
}

// Scalar tail for rows not covered by 16-row tiles (N % 16 != 0 only).
__global__ void k_gemm1_tail(const float* __restrict__ x,
                             const float* __restrict__ W1,
                             const float* __restrict__ dinv,
                             float* __restrict__ hs,
                             float* __restrict__ agg1,
                             int row_start, int n) {
    int row = row_start + blockIdx.x * blockDim.x + threadIdx.x;
    if (row >= n) return;
    float di = dinv[row];
    const float* xr = x + (size_t)row * D_IN;
    for (int j = 0; j < D_HID; ++j) {
        float acc = 0.0f;
        for (int k = 0; k < D_IN; ++k) acc = fmaf(xr[k], W1[k * D_HID + j], acc);
        float hv = acc * di;
        hs[(size_t)row * D_HID + j]   = hv;
        agg1[(size_t)row * D_HID + j] = hv * di;
    }
}

// ---------------------------------------------------------------------------
// Kernel 5: layer-1 scatter: agg1[dst,k] += hs[src,k] * dinv[dst]
// one thread per (edge, k): 16 consecutive threads read 64B of hs[src] coalesced
__global__ void k_edges1(const int* __restrict__ src, const int* __restrict__ dst,
                         const float* __restrict__ dinv,
                         const float* __restrict__ hs,
                         float* __restrict__ agg1, int e) {
    long long t = (long long)blockIdx.x * blockDim.x + threadIdx.x;
    if (t >= (long long)e * D_HID) return;
    int ei = (int)(t >> 4);
    int k  = (int)(t & 15);
    int s = src[ei], d = dst[ei];
    float w = dinv[d];
    atomicAdd(&agg1[(size_t)d * D_HID + k], hs[(size_t)s * D_HID + k] * w);
}

// ---------------------------------------------------------------------------
// Kernel 6: per node: r = relu(agg1 + b1); h2 = r @ W2;
//   hs2  = h2 * dinv;   agg2 = hs2 * dinv (self-loop seed)
__global__ void k_layer2_node(const float* __restrict__ agg1,
                              const float* __restrict__ b1,
                              const float* __restrict__ W2,
                              const float* __restrict__ dinv,
                              float* __restrict__ hs2,
                              float* __restrict__ agg2, int n) {
    int i = blockIdx.x * blockDim.x + threadIdx.x;
    if (i >= n) return;
    float di = dinv[i];
    float acc0 = 0.0f, acc1 = 0.0f;
    const float* a = agg1 + (size_t)i * D_HID;
    #pragma unroll
    for (int k = 0; k < D_HID; ++k) {
        float r = a[k] + b1[k];
        r = r > 0.0f ? r : 0.0f;
        acc0 = fmaf(r, W2[k * D_OUT + 0], acc0);
        acc1 = fmaf(r, W2[k * D_OUT + 1], acc1);
    }
    float h0 = acc0 * di, h1 = acc1 * di;
    hs2[(size_t)i * D_OUT + 0]  = h0;
    hs2[(size_t)i * D_OUT + 1]  = h1;
    agg2[(size_t)i * D_OUT + 0] = h0 * di;
    agg2[(size_t)i * D_OUT + 1] = h1 * di;
}

// Kernel 7: layer-2 scatter: agg2[dst,c] += hs2[src,c] * dinv[dst]
__global__ void k_edges2(const int* __restrict__ src, const int* __restrict__ dst,
                         const float* __restrict__ dinv,
                         const float* __restrict__ hs2,
                         float* __restrict__ agg2, int e) {
    long long t = (long long)blockIdx.x * blockDim.x + threadIdx.x;
    if (t >= (long long)e * D_OUT) return;
    int ei = (int)(t >> 1);
    int c  = (int)(t & 1);
    int s = src[ei], d = dst[ei];
    atomicAdd(&agg2[(size_t)d * D_OUT + c], hs2[(size_t)s * D_OUT + c] * dinv[d]);
}

// Kernel 8: out = log_softmax(agg2 + b2) over the 2 classes (stable)
__global__ void k_finalize(const float* __restrict__ agg2,
                           const float* __restrict__ b2,
                           float* __restrict__ out, int n) {
    int i = blockIdx.x * blockDim.x + threadIdx.x;
    if (i >= n) return;
    float v0 = agg2[(size_t)i * 2 + 0] + b2[0];
    float v1 = agg2[(size_t)i * 2 + 1] + b2[1];
    float m  = fmaxf(v0, v1);
    float lse = m + logf(expf(v0 - m) + expf(v1 - m));
    out[(size_t)i * 2 + 0] = v0 - lse;
    out[(size_t)i * 2 + 1] = v1 - lse;
}

// ---------------------------------------------------------------------------
extern "C" void kernel_launch(void* const* d_in, const int* in_sizes, int n_in,
                              void* d_out, int out_size, void* d_ws, size_t ws_size,
                              hipStream_t stream) {
    const float* x  = (const float*)d_in[0];
    const float* W1 = (const float*)d_in[1];
    const float* b1 = (const float*)d_in[2];
    const float* W2 = (const float*)d_in[3];
    const float* b2 = (const float*)d_in[4];
    const int*   ei = (const int*)d_in[5];
    float* out = (float*)d_out;

    const int N = in_sizes[0] / D_IN;     // 100000
    const int E = in_sizes[5] / 2;        // 3200000
    const int* src = ei;                  // edge_index[0]
    const int* dst = ei + E;              // edge_index[1]

    // workspace layout (floats): dinv[N] | hs[N*16] | agg1[N*16] | hs2[N*2] | agg2[N*2]
    float* wsf  = (float*)d_ws;
    float* dinv = wsf;
    float* hs   = wsf + (size_t)N;
    float* agg1 = wsf + (size_t)N * 17;
    float* hs2  = wsf + (size_t)N * 33;
    float* agg2 = wsf + (size_t)N * 35;

    const int B = 256;

    // degree -> dinv
    k_deg_init  <<<(N + B - 1) / B, B, 0, stream>>>(dinv, N);
    k_deg_accum <<<(E + B - 1) / B, B, 0, stream>>>(dst, dinv, E);
    k_deg_rsqrt <<<(N + B - 1) / B, B, 0, stream>>>(dinv, N);

    // layer 1 GEMM (WMMA fp32, one wave per 16-row tile; 4 waves / block)
    int ntiles = N / 16;
    if (ntiles > 0) {
        int wavesPerBlock = 4;
        int nblk = (ntiles + wavesPerBlock - 1) / wavesPerBlock;
        k_gemm1_wmma<<<nblk, wavesPerBlock * 32, 0, stream>>>(x, W1, dinv, hs, agg1, ntiles);
    }
    int tail = N - ntiles * 16;
    if (tail > 0) {
        k_gemm1_tail<<<(tail + B - 1) / B, B, 0, stream>>>(x, W1, dinv, hs, agg1, ntiles * 16, N);
    }

    // layer 1 edge scatter
    {
        long long work = (long long)E * D_HID;
        long long nblk = (work + B - 1) / B;
        k_edges1<<<(unsigned)nblk, B, 0, stream>>>(src, dst, dinv, hs, agg1, E);
    }

    // layer 2 per-node transform
    k_layer2_node<<<(N + B - 1) / B, B, 0, stream>>>(agg1, b1, W2, dinv, hs2, agg2, N);

    // layer 2 edge scatter
    {
        long long work = (long long)E * D_OUT;
        long long nblk = (work + B - 1) / B;
        k_edges2<<<(unsigned)nblk, B, 0, stream>>>(src, dst, dinv, hs2, agg2, E);
    }

    // bias + log_softmax
    k_finalize<<<(N + B - 1) / B, B, 0, stream>>>(agg2, b2, out, N);
}